// MeshConv_43928925503801
// MI455X (gfx1250) — compile-verified
//
#include <hip/hip_runtime.h>
#include <hip/hip_bf16.h>

typedef __attribute__((ext_vector_type(2))) float v2f;
typedef __attribute__((ext_vector_type(8))) float v8f;

#define C_DIM 128          // C_IN == C_OUT == 128
#define WAVES_PER_BLOCK 8  // 256 threads = 8 wave32

// ---------------------------------------------------------------------------
// Phase 1: COO scatter  agg[rows[e]] += vals[e] * x[cols[e]]
// One wave32 per edge: lane L covers channels 4L..4L+3 (float4 gather,
// 4x global_atomic_add_f32 into L2-resident agg).
// ---------------------------------------------------------------------------
__global__ void __launch_bounds__(256)
spmm_scatter_kernel(const float* __restrict__ x,
                    const int*   __restrict__ rows,
                    const int*   __restrict__ cols,
                    const float* __restrict__ vals,
                    float*       __restrict__ agg,
                    int n_edges) {
    int wave = (int)((blockIdx.x * blockDim.x + threadIdx.x) >> 5);
    int lane = (int)(threadIdx.x & 31);
    if (wave >= n_edges) return;           // wave-uniform guard

    int   r = rows[wave];
    int   c = cols[wave];
    float v = vals[wave];

    const float4* xrow = (const float4*)(x + (size_t)c * C_DIM);
    float4 xv = xrow[lane];                // global_load_b128, L2 hit for hot nodes

    float* dst = agg + (size_t)r * C_DIM + lane * 4;
    atomicAdd(dst + 0, v * xv.x);
    atomicAdd(dst + 1, v * xv.y);
    atomicAdd(dst + 2, v * xv.z);
    atomicAdd(dst + 3, v * xv.w);
}

// ---------------------------------------------------------------------------
// Phase 2: out = agg @ W^T + b  via V_WMMA_F32_16X16X4_F32 (fp32 matrix core).
// Each wave computes a 16(node) x 128(out-channel) strip: 8 accumulator tiles,
// K-loop of 32 steps (K=4 per WMMA). A-fragment: one float2 per lane per step.
// B-fragment: W row slice, 64 KB total -> L0/L2 resident.
//
// A 16x4 f32 layout : VGPR0/1, lanes 0-15 hold K=0/1, lanes 16-31 hold K=2/3,
//                     M = lane & 15.
// B 4x16 f32 layout : mirrored, N = lane & 15, same K split.
// D 16x16 f32 layout: VGPR r -> M = r + 8*(lane>=16), N = lane & 15.
// ---------------------------------------------------------------------------
__global__ void __launch_bounds__(256)
gemm_wmma_kernel(const float* __restrict__ agg,
                 const float* __restrict__ W,   // [C_OUT, C_IN] row-major
                 const float* __restrict__ bias,
                 float*       __restrict__ out,
                 int n_tiles) {               // n_tiles = n_nodes / 16
    int wave = (int)(threadIdx.x >> 5);
    int lane = (int)(threadIdx.x & 31);
    int tile = (int)blockIdx.x * WAVES_PER_BLOCK + wave;
    if (tile >= n_tiles) return;            // wave-uniform: EXEC all-1 below

    int m0   = tile * 16;
    int nrow = lane & 15;                   // M (A) / N (B,D) within tile
    int koff = (lane >> 4) * 2;             // K sub-offset for this half-wave

    const float* arow = agg + (size_t)(m0 + nrow) * C_DIM + koff;

    v8f acc[8];
    #pragma unroll
    for (int j = 0; j < 8; ++j) acc[j] = (v8f){0.f,0.f,0.f,0.f,0.f,0.f,0.f,0.f};

    for (int k0 = 0; k0 < C_DIM; k0 += 4) {
        v2f a = *(const v2f*)(arow + k0);   // A fragment: agg rows, streamed once
        #pragma unroll
        for (int j = 0; j < 8; ++j) {
            // B fragment: W^T tile -> lane reads W[16j + nrow][k0+koff .. +1]
            const float* wrow = W + (size_t)(16 * j + nrow) * C_DIM + k0 + koff;
            v2f bb = *(const v2f*)wrow;
            acc[j] = __builtin_amdgcn_wmma_f32_16x16x4_f32(
                false, a, false, bb, (short)0, acc[j], false, false);
        }
    }

    // Epilogue: bias + store. D-layout: M = r + 8*(lane>=16), N = nrow.
    int mbase = m0 + (lane >> 4) * 8;
    #pragma unroll
    for (int j = 0; j < 8; ++j) {
        float bv = bias[16 * j + nrow];
        #pragma unroll
        for (int r = 0; r < 8; ++r) {
            out[(size_t)(mbase + r) * C_DIM + 16 * j + nrow] = acc[j][r] + bv;
        }
    }
}

// ---------------------------------------------------------------------------
// Launch: memset(agg) -> scatter -> WMMA GEMM, all on `stream`.
// d_in order: x, rows, cols, vals, W, b
// ---------------------------------------------------------------------------
extern "C" void kernel_launch(void* const* d_in, const int* in_sizes, int n_in,
                              void* d_out, int out_size, void* d_ws, size_t ws_size,
                              hipStream_t stream) {
    const float* x    = (const float*)d_in[0];
    const int*   rows = (const int*)  d_in[1];
    const int*   cols = (const int*)  d_in[2];
    const float* vals = (const float*)d_in[3];
    const float* W    = (const float*)d_in[4];
    const float* bias = (const float*)d_in[5];
    float* out = (float*)d_out;

    int n_nodes = in_sizes[0] / C_DIM;   // 100000
    int n_edges = in_sizes[1];           // 1600000

    float* agg = (float*)d_ws;           // n_nodes * 128 floats = 51.2 MB
    size_t agg_bytes = (size_t)n_nodes * C_DIM * sizeof(float);

    // Zero accumulator (graph-capture-legal stream memset).
    hipMemsetAsync(agg, 0, agg_bytes, stream);

    // Phase 1: one wave per edge, 8 edges per 256-thread block.
    int blocks1 = (n_edges + WAVES_PER_BLOCK - 1) / WAVES_PER_BLOCK;
    spmm_scatter_kernel<<<blocks1, 256, 0, stream>>>(x, rows, cols, vals, agg, n_edges);

    // Phase 2: one wave per 16-node tile, 8 tiles per block.
    int n_tiles = n_nodes / 16;          // 6250 (exact)
    int blocks2 = (n_tiles + WAVES_PER_BLOCK - 1) / WAVES_PER_BLOCK;
    gemm_wmma_kernel<<<blocks2, 256, 0, stream>>>(agg, W, bias, out, n_tiles);
}